// spuLayerTransformer_4595615007368
// MI455X (gfx1250) — compile-verified
//
#include <hip/hip_runtime.h>

// Elementwise SPU bound-propagation kernel for MI455X (gfx1250, wave32).
// Memory: 28 B/elem -> ~10us floor at 23.3 TB/s (b128 NT loads/stores).
// Compute: trimmed to ~9 trans (4 v_exp_f32 + 5 v_rcp_f32) and ~110 VALU
// per element by (a) v_rcp_f32 instead of full-precision divides,
// (b) cancelling the secant side of every span out of the argmin,
// (c) reusing endpoint sigmoids for the clipped heuristic points,
// (d) v_med3_f32 clips and closed-form tangent intercepts.

typedef float v4f __attribute__((ext_vector_type(4)));

__device__ __forceinline__ float rcp_fast(float v) {
  return __builtin_amdgcn_rcpf(v);  // v_rcp_f32, ~1 ulp
}

__device__ __forceinline__ float sig_neg_fast(float v) {
  // sigmoid(-v) = 1 / (1 + e^v):  v_mul(log2e) + v_exp_f32 + v_add + v_rcp
  return rcp_fast(1.0f + __expf(v));
}

// Argmin key for heuristic point p. The secant-side term of the span is
// common to all three points and cancels from the argmin:
//   u>0 : span = C - lo(p)  -> key = -lo(p)
//   u<=0: span = hi(p) - C  -> key =  hi(p)
__device__ __forceinline__ float span_key(float p, float sg, float l, float u,
                                          float l0_s, bool upos) {
  bool  pp     = (p >= 0.0f);
  float s_pos  = p + p;                         // dx_spu(p), p>=0
  float i_pos  = fmaf(-p, p, -0.5f);            // spu(p)-p*dx = -p^2-0.5
  float ts_neg = fmaf(sg, sg, -sg);             // -sg*(1-sg), p<0
  float i_neg  = fmaf(-p, ts_neg, sg - 1.0f);   // (sg-1) - p*ts
  float tan_s  = pp ? s_pos : ts_neg;
  float tan_i  = pp ? i_pos : i_neg;

  // active (non-secant) bound line for this point
  float s = upos ? (pp ? s_pos : l0_s) : tan_s;
  float i = upos ? (pp ? i_pos : -0.5f) : tan_i;

  // concretize: lower bound (upos) picks l if s>0 else u; upper picks u/l
  bool  sel  = (s > 0.0f);
  float pick = (sel == upos) ? l : u;
  float val  = fmaf(pick, s, i);
  return upos ? -val : val;
}

__device__ __forceinline__ void spu_elem(float x, float l, float u,
                                         float& xo, float& lo_out,
                                         float& uo_out, float& ho) {
  // endpoint + x sigmoids (the only ones that need exp besides mid)
  float sg_x = sig_neg_fast(x);
  float sg_l = sig_neg_fast(l);
  float sg_u = sig_neg_fast(u);

  float sl = (l >= 0.0f) ? fmaf(l, l, -0.5f) : (sg_l - 1.0f);
  float su = (u >= 0.0f) ? fmaf(u, u, -0.5f) : (sg_u - 1.0f);
  xo       = (x >= 0.0f) ? fmaf(x, x, -0.5f) : (sg_x - 1.0f);

  // box bounds
  bool lpos = (l >= 0.0f);
  bool uneg = (u <= 0.0f);
  lo_out = lpos ? sl : (uneg ? su : -0.5f);
  uo_out = lpos ? su : (uneg ? sl : fmaxf(sl, su));

  // line through (l, spu(l)) and (0, -0.5)
  float l0_d = (l != 0.0f) ? -l : 1.0f;
  float l0_s = (-0.5f - sl) * rcp_fast(l0_d);
  bool  upos = (u > 0.0f);

  // heuristic tangent points: clip(x), clip(0), clip(mid)  (v_med3_f32)
  float mid = fmaf(u - l, 0.5f, l);
  float p0  = __builtin_amdgcn_fmed3f(x, l, u);
  float p1  = __builtin_amdgcn_fmed3f(0.0f, l, u);
  float p2  = __builtin_amdgcn_fmed3f(mid, l, u);

  // sigmoid(-p): reuse endpoint sigmoids for p0/p1, compute only for mid
  float sg_p0 = (x < l) ? sg_l : ((x > u) ? sg_u : sg_x);
  float sg_p1 = (l > 0.0f) ? sg_l : ((u < 0.0f) ? sg_u : 0.5f);
  float sg_p2 = sig_neg_fast(mid);

  float k0 = span_key(p0, sg_p0, l, u, l0_s, upos);
  float k1 = span_key(p1, sg_p1, l, u, l0_s, upos);
  float k2 = span_key(p2, sg_p2, l, u, l0_s, upos);

  // first-min argmin (jnp/torch tie-break)
  int   idx  = 0;
  float best = k0;
  if (k1 < best) { best = k1; idx = 1; }
  if (k2 < best) { idx = 2; }
  ho = (float)idx;
}

__global__ __launch_bounds__(256) void spu_v4_kernel(
    const float* __restrict__ x, const float* __restrict__ l,
    const float* __restrict__ u, float* __restrict__ out, int n) {
  int i4 = blockIdx.x * blockDim.x + threadIdx.x;
  int n4 = n >> 2;
  if (i4 >= n4) return;

  v4f xv = __builtin_nontemporal_load((const v4f*)x + i4);
  v4f lv = __builtin_nontemporal_load((const v4f*)l + i4);
  v4f uv = __builtin_nontemporal_load((const v4f*)u + i4);

  v4f xo, lo, uo, ho;
#pragma unroll
  for (int k = 0; k < 4; ++k) {
    float a, b, c, h;
    spu_elem(xv[k], lv[k], uv[k], a, b, c, h);
    xo[k] = a; lo[k] = b; uo[k] = c; ho[k] = h;
  }

  __builtin_nontemporal_store(xo, (v4f*)(out)         + i4);
  __builtin_nontemporal_store(lo, (v4f*)(out + n)     + i4);
  __builtin_nontemporal_store(uo, (v4f*)(out + 2 * n) + i4);
  __builtin_nontemporal_store(ho, (v4f*)(out + 3 * n) + i4);
}

// Generic scalar fallback (n not divisible by 4).
__global__ __launch_bounds__(256) void spu_scalar_kernel(
    const float* __restrict__ x, const float* __restrict__ l,
    const float* __restrict__ u, float* __restrict__ out, int n) {
  int i = blockIdx.x * blockDim.x + threadIdx.x;
  if (i >= n) return;
  float a, b, c, h;
  spu_elem(x[i], l[i], u[i], a, b, c, h);
  out[i]         = a;
  out[n + i]     = b;
  out[2 * n + i] = c;
  out[3 * n + i] = h;
}

extern "C" void kernel_launch(void* const* d_in, const int* in_sizes, int n_in,
                              void* d_out, int out_size, void* d_ws,
                              size_t ws_size, hipStream_t stream) {
  (void)n_in; (void)out_size; (void)d_ws; (void)ws_size;
  const float* x = (const float*)d_in[0];
  const float* l = (const float*)d_in[1];
  const float* u = (const float*)d_in[2];
  float* out = (float*)d_out;
  int n = in_sizes[0];
  if (n <= 0) return;

  const int block = 256;  // 8 wave32s
  if ((n & 3) == 0) {
    int n4 = n >> 2;
    int grid = (n4 + block - 1) / block;  // 8192 blocks for n = 2^23
    spu_v4_kernel<<<grid, block, 0, stream>>>(x, l, u, out, n);
  } else {
    int grid = (n + block - 1) / block;
    spu_scalar_kernel<<<grid, block, 0, stream>>>(x, l, u, out, n);
  }
}